// ClusterAssignerVoxelSPCCL_10428180594807
// MI455X (gfx1250) — compile-verified
//
#include <hip/hip_runtime.h>
#include <stdint.h>

#define GMAXD 1024
#define VMAXD 10240
#define ITERSD 12
#define MIN_PTS 3
#define BSZD 2
#define HCAP 32768
#define SORT_N 16384
#define NWORDS (VMAXD / 32) /* 320 */
#define EMPTYKEY 0xFFFFFFFFFFFFFFFFull
#define FILLKEY (1ull << 62)

typedef float v2f __attribute__((ext_vector_type(2)));
typedef float v8f __attribute__((ext_vector_type(8)));

__device__ __forceinline__ uint32_t hash64(unsigned long long k) {
  k *= 0x9E3779B97F4A7C15ull;
  return (uint32_t)(k >> 32) & (HCAP - 1);
}

__device__ __forceinline__ unsigned long long voxkey(float px, float py, float pz,
                                                     const float* vcfg, int c, int b,
                                                     int* gxo, int* gyo, int* gzo) {
  float ox = vcfg[c * 6 + 0], oy = vcfg[c * 6 + 1], oz = vcfg[c * 6 + 2];
  float sx = vcfg[c * 6 + 3], sy = vcfg[c * 6 + 4], sz = vcfg[c * 6 + 5];
  int gx = (int)floorf((px - ox) / sx);
  int gy = (int)floorf((py - oy) / sy);
  int gz = (int)floorf((pz - oz) / sz);
  gx = gx < 0 ? 0 : (gx > GMAXD - 1 ? GMAXD - 1 : gx);
  gy = gy < 0 ? 0 : (gy > GMAXD - 1 ? GMAXD - 1 : gy);
  gz = gz < 0 ? 0 : (gz > GMAXD - 1 ? GMAXD - 1 : gz);
  *gxo = gx; *gyo = gy; *gzo = gz;
  long long key = (long long)(c * BSZD + b);
  key = ((key * GMAXD + gx) * GMAXD + gy) * GMAXD + gz;
  return (unsigned long long)key;
}

__global__ void k_init(unsigned long long* hkey, int* hval, int* ucnt,
                       float* vx, float* vy, float* vz, float* vsq, float* vthr2,
                       int* vcls, int* vbat, int* labels, int* counts) {
  int t = blockIdx.x * blockDim.x + threadIdx.x;
  if (t < HCAP) { hkey[t] = EMPTYKEY; hval[t] = -1; }
  if (t < VMAXD) {
    vx[t] = 0.f; vy[t] = 0.f; vz[t] = 0.f; vsq[t] = 0.f; vthr2[t] = 0.f;
    vcls[t] = -1; vbat[t] = -1; labels[t] = t; counts[t] = 0;
  }
  if (t == 0) *ucnt = 0;
}

__global__ void k_build(const float* pts, const float* vcfg, const int* bat, const int* cls,
                        int n, unsigned long long* hkey, int* ucnt, unsigned long long* uniq) {
  int p = blockIdx.x * blockDim.x + threadIdx.x;
  if (p >= n) return;
  int c = cls[p], b = bat[p], gx, gy, gz;
  unsigned long long k = voxkey(pts[p * 3 + 0], pts[p * 3 + 1], pts[p * 3 + 2], vcfg, c, b, &gx, &gy, &gz);
  uint32_t h = hash64(k);
  for (;;) {
    unsigned long long prev = atomicCAS(&hkey[h], EMPTYKEY, k);
    if (prev == EMPTYKEY) {
      int idx = atomicAdd(ucnt, 1);
      if (idx < SORT_N) uniq[idx] = k;
      break;
    }
    if (prev == k) break;
    h = (h + 1) & (HCAP - 1);
  }
}

extern __shared__ unsigned long long skey[];
__global__ void k_sort(const unsigned long long* uniq, const int* ucnt,
                       const unsigned long long* hkey, int* hval) {
  int t = threadIdx.x; /* 1024 threads */
  int U = *ucnt; if (U > SORT_N) U = SORT_N;
  for (int i = t; i < SORT_N; i += 1024) skey[i] = (i < U) ? uniq[i] : FILLKEY;
  __syncthreads();
  for (int k = 2; k <= SORT_N; k <<= 1) {
    for (int j = k >> 1; j > 0; j >>= 1) {
      for (int i = t; i < SORT_N; i += 1024) {
        int ixj = i ^ j;
        if (ixj > i) {
          bool up = ((i & k) == 0);
          unsigned long long a = skey[i], b = skey[ixj];
          if ((a > b) == up) { skey[i] = b; skey[ixj] = a; }
        }
      }
      __syncthreads();
    }
  }
  for (int s = t; s < VMAXD; s += 1024) {
    unsigned long long key = skey[s];
    if (key == FILLKEY) continue;
    uint32_t h = hash64(key);
    while (hkey[h] != key) h = (h + 1) & (HCAP - 1);
    hval[h] = s;
  }
}

__global__ void k_scatter(const float* pts, const float* vcfg, const float* cdist,
                          const int* bat, const int* cls, int n,
                          const unsigned long long* hkey, const int* hval,
                          int* inv, float* vx, float* vy, float* vz, float* vsq,
                          float* vthr2, int* vcls, int* vbat) {
  int p = blockIdx.x * blockDim.x + threadIdx.x;
  if (p >= n) return;
  int c = cls[p], b = bat[p], gx, gy, gz;
  unsigned long long k = voxkey(pts[p * 3 + 0], pts[p * 3 + 1], pts[p * 3 + 2], vcfg, c, b, &gx, &gy, &gz);
  uint32_t h = hash64(k);
  while (hkey[h] != k) h = (h + 1) & (HCAP - 1);
  int s = hval[h];
  if (s < 0 || s >= VMAXD) s = VMAXD - 1;
  inv[p] = s;
  float cx = vcfg[c * 6 + 0] + ((float)gx + 0.5f) * vcfg[c * 6 + 3];
  float cy = vcfg[c * 6 + 1] + ((float)gy + 0.5f) * vcfg[c * 6 + 4];
  float cz = vcfg[c * 6 + 2] + ((float)gz + 0.5f) * vcfg[c * 6 + 5];
  vx[s] = cx; vy[s] = cy; vz[s] = cz;
  vsq[s] = cx * cx + cy * cy + cz * cz;
  vcls[s] = c; vbat[s] = b;
  float cd = cdist[c];
  vthr2[s] = cd * cd;
}

/* Adjacency via V_WMMA_F32_16X16X4_F32: dot = vcen @ vcen.T with K=3 padded to 4.
   Each wave: 16 rows x 32 cols -> two WMMAs -> one 32-bit bitmask word per row. */
__global__ void k_adj(const float* __restrict__ vx, const float* __restrict__ vy,
                      const float* __restrict__ vz, const float* __restrict__ vsq,
                      const float* __restrict__ vthr2, const int* __restrict__ vcls,
                      const int* __restrict__ vbat, unsigned int* __restrict__ adj) {
  int lane = threadIdx.x & 31;
  int wave = threadIdx.x >> 5;
  int jb = blockIdx.x * 8 + wave; /* column word index, 0..319 */
  int i0 = blockIdx.y * 16;       /* row tile base */
  int m = lane & 15;
  int hi = lane >> 4;             /* lanes 16..31 carry K=2 (z) / K=3 (0) */
  int row = i0 + m;
  v2f a, b0, b1;
  a.x = hi ? vz[row] : vx[row];
  a.y = hi ? 0.f : vy[row];
  int colA = jb * 32 + m, colB = colA + 16;
  b0.x = hi ? vz[colA] : vx[colA];
  b0.y = hi ? 0.f : vy[colA];
  b1.x = hi ? vz[colB] : vx[colB];
  b1.y = hi ? 0.f : vy[colB];
  v8f cacc = {0.f, 0.f, 0.f, 0.f, 0.f, 0.f, 0.f, 0.f};
  v8f d0 = __builtin_amdgcn_wmma_f32_16x16x4_f32(false, a, false, b0, (short)0, cacc, false, false);
  v8f d1 = __builtin_amdgcn_wmma_f32_16x16x4_f32(false, a, false, b1, (short)0, cacc, false, false);
  int clsA = vcls[colA], clsB = vcls[colB];
  int batA = vbat[colA], batB = vbat[colB];
  float sqA = vsq[colA], sqB = vsq[colB];
  unsigned words[16];
#pragma unroll
  for (int k = 0; k < 8; ++k) {
    int r = i0 + k + 8 * hi; /* D layout: lanes16-31 hold M = 8+k */
    int rc = vcls[r], rb = vbat[r];
    float rs = vsq[r], rt = vthr2[r];
    float d2a = rs + sqA - 2.f * d0[k];
    float d2b = rs + sqB - 2.f * d1[k];
    bool pa = (rc >= 0) && (rc == clsA) && (rb == batA) && (d2a <= rt);
    bool pb = (rc >= 0) && (rc == clsB) && (rb == batB) && (d2b <= rt);
    unsigned m0 = (unsigned)__ballot(pa);
    unsigned m1 = (unsigned)__ballot(pb);
    words[k] = (m0 & 0xFFFFu) | (m1 << 16);
    words[k + 8] = (m0 >> 16) | (m1 & 0xFFFF0000u);
  }
  if (lane == 0) {
#pragma unroll
    for (int k = 0; k < 16; ++k)
      adj[(size_t)(i0 + k) * NWORDS + jb] = words[k];
  }
}

__global__ void k_prop_a(const unsigned int* __restrict__ adj,
                         const int* __restrict__ labels, int* __restrict__ nl) {
  __shared__ int slab[VMAXD];
  for (int i = threadIdx.x; i < VMAXD; i += blockDim.x) slab[i] = labels[i];
  __syncthreads();
  int lane = threadIdx.x & 31;
  int wave = threadIdx.x >> 5;
  int r = blockIdx.x * 8 + wave;
  const unsigned int* rowp = adj + (size_t)r * NWORDS;
  __builtin_prefetch(rowp, 0, 1);
  int best = 0x7FFFFFFF;
  for (int t = lane; t < NWORDS; t += 32) {
    unsigned w = rowp[t];
    while (w) {
      int b = __ffs(w) - 1;
      w &= w - 1;
      int lbl = slab[t * 32 + b];
      best = best < lbl ? best : lbl;
    }
  }
  for (int off = 16; off; off >>= 1) {
    int o = __shfl_xor(best, off, 32);
    best = best < o ? best : o;
  }
  if (lane == 0) {
    int cur = slab[r];
    nl[r] = cur < best ? cur : best;
  }
}

__global__ void k_prop_b(const int* __restrict__ nl, int* __restrict__ labels) {
  int v = blockIdx.x * blockDim.x + threadIdx.x;
  if (v < VMAXD) {
    int l = nl[v];
    labels[v] = nl[l]; /* pointer jump: labels = L[L] */
  }
}

__global__ void k_counts(const int* inv, const int* labels, int n, int* counts) {
  int p = blockIdx.x * blockDim.x + threadIdx.x;
  if (p >= n) return;
  atomicAdd(&counts[labels[inv[p]]], 1);
}

__global__ void k_scan(const int* counts, int* cid) {
  __shared__ int part[1024];
  int t = threadIdx.x;
  const int CH = VMAXD / 1024; /* 10 */
  int base = t * CH;
  int s = 0;
  for (int i = 0; i < CH; ++i) s += (counts[base + i] >= MIN_PTS) ? 1 : 0;
  part[t] = s;
  __syncthreads();
  for (int off = 1; off < 1024; off <<= 1) {
    int v = (t >= off) ? part[t - off] : 0;
    __syncthreads();
    part[t] += v;
    __syncthreads();
  }
  int run = (t == 0) ? 0 : part[t - 1];
  for (int i = 0; i < CH; ++i) {
    run += (counts[base + i] >= MIN_PTS) ? 1 : 0;
    cid[base + i] = run - 1;
  }
}

__global__ void k_out(const int* cls, const int* bat, const int* inv,
                      const int* labels, const int* counts, const int* cid,
                      int n, float* out) {
  int p = blockIdx.x * blockDim.x + threadIdx.x;
  if (p >= n) return;
  int v = inv[p];
  int root = labels[v];
  int cnt = counts[root];
  bool valid = cnt >= MIN_PTS;
  int cluster = valid ? cid[root] : -1;
  out[3 * p + 0] = (float)cls[p];
  out[3 * p + 1] = (float)bat[p];
  out[3 * p + 2] = (float)cluster;
  out[3 * (size_t)n + p] = valid ? 1.f : 0.f;
}

static inline size_t alignup(size_t x) { return (x + 255) & ~(size_t)255; }

extern "C" void kernel_launch(void* const* d_in, const int* in_sizes, int n_in,
                              void* d_out, int out_size, void* d_ws, size_t ws_size,
                              hipStream_t stream) {
  const float* points = (const float*)d_in[0];
  const float* vcfg = (const float*)d_in[1];
  const float* cdist = (const float*)d_in[2];
  const int* batch_id = (const int*)d_in[3];
  const int* class_id = (const int*)d_in[4];
  int N = in_sizes[0] / 3;

  char* ws = (char*)d_ws;
  size_t off = 0;
  unsigned long long* hkey = (unsigned long long*)(ws + off); off = alignup(off + (size_t)HCAP * 8);
  int* hval = (int*)(ws + off);                               off = alignup(off + (size_t)HCAP * 4);
  unsigned long long* uniq = (unsigned long long*)(ws + off); off = alignup(off + (size_t)SORT_N * 8);
  int* ucnt = (int*)(ws + off);                               off = alignup(off + 256);
  float* vx = (float*)(ws + off);                             off = alignup(off + (size_t)VMAXD * 4);
  float* vy = (float*)(ws + off);                             off = alignup(off + (size_t)VMAXD * 4);
  float* vz = (float*)(ws + off);                             off = alignup(off + (size_t)VMAXD * 4);
  float* vsq = (float*)(ws + off);                            off = alignup(off + (size_t)VMAXD * 4);
  float* vthr2 = (float*)(ws + off);                          off = alignup(off + (size_t)VMAXD * 4);
  int* vcls = (int*)(ws + off);                               off = alignup(off + (size_t)VMAXD * 4);
  int* vbat = (int*)(ws + off);                               off = alignup(off + (size_t)VMAXD * 4);
  int* labels = (int*)(ws + off);                             off = alignup(off + (size_t)VMAXD * 4);
  int* nl = (int*)(ws + off);                                 off = alignup(off + (size_t)VMAXD * 4);
  int* counts = (int*)(ws + off);                             off = alignup(off + (size_t)VMAXD * 4);
  int* cid = (int*)(ws + off);                                off = alignup(off + (size_t)VMAXD * 4);
  int* inv = (int*)(ws + off);                                off = alignup(off + (size_t)N * 4);
  unsigned int* adj = (unsigned int*)(ws + off);              off = alignup(off + (size_t)VMAXD * NWORDS * 4);
  (void)ws_size; (void)out_size; (void)n_in;

  int nb = (N + 255) / 256;

  k_init<<<HCAP / 256, 256, 0, stream>>>(hkey, hval, ucnt, vx, vy, vz, vsq, vthr2,
                                         vcls, vbat, labels, counts);
  k_build<<<nb, 256, 0, stream>>>(points, vcfg, batch_id, class_id, N, hkey, ucnt, uniq);
  k_sort<<<1, 1024, (size_t)SORT_N * 8, stream>>>(uniq, ucnt, hkey, hval);
  k_scatter<<<nb, 256, 0, stream>>>(points, vcfg, cdist, batch_id, class_id, N,
                                    hkey, hval, inv, vx, vy, vz, vsq, vthr2, vcls, vbat);
  k_adj<<<dim3(NWORDS / 8, VMAXD / 16), 256, 0, stream>>>(vx, vy, vz, vsq, vthr2, vcls, vbat, adj);
  for (int it = 0; it < ITERSD; ++it) {
    k_prop_a<<<VMAXD / 8, 256, 0, stream>>>(adj, labels, nl);
    k_prop_b<<<VMAXD / 256, 256, 0, stream>>>(nl, labels);
  }
  k_counts<<<nb, 256, 0, stream>>>(inv, labels, N, counts);
  k_scan<<<1, 1024, 0, stream>>>(counts, cid);
  k_out<<<nb, 256, 0, stream>>>(class_id, batch_id, inv, labels, counts, cid, N, (float*)d_out);
}